// SimpleGNN_83958020702803
// MI455X (gfx1250) — compile-verified
//
#include <hip/hip_runtime.h>
#include <cstdint>

#define NN     50000
#define FIN    128
#define H1DIM  100
#define H2DIM  200
#define FOUT   16

typedef float v2f __attribute__((ext_vector_type(2)));
typedef float v8f __attribute__((ext_vector_type(8)));

// ---------------- degree / normalization ----------------

__global__ void k_fill(float* __restrict__ p, float v, int n) {
    int i = blockIdx.x * blockDim.x + threadIdx.x;
    if (i < n) p[i] = v;
}

__global__ void k_deg(const int* __restrict__ col, float* __restrict__ deg, int E) {
    int e = blockIdx.x * blockDim.x + threadIdx.x;
    if (e < E) atomicAdd(&deg[col[e]], 1.0f);
}

__global__ void k_dis(const float* __restrict__ deg, float* __restrict__ dis, int n) {
    int i = blockIdx.x * blockDim.x + threadIdx.x;
    if (i < n) dis[i] = __frsqrt_rn(deg[i]);   // deg >= 1 always (self loop)
}

__global__ void k_norm(const int* __restrict__ row, const int* __restrict__ col,
                       const float* __restrict__ dis, float* __restrict__ nrm, int E) {
    int e = blockIdx.x * blockDim.x + threadIdx.x;
    if (e < E) nrm[e] = dis[row[e]] * dis[col[e]];
}

// ---------------- WMMA f32 GEMM: C[M,N] = op(A[M,K]) @ B[K,N] (+bias) ----------------
// One wave per 16x16 output tile. M % 16 == 0, K % 4 == 0 guaranteed by caller.
// N tail handled by value-select (EXEC stays all-ones at the WMMA).

template <bool RELU_A, bool ADD_BIAS>
__global__ void k_gemm_wmma(const float* __restrict__ A, const float* __restrict__ B,
                            const float* __restrict__ bias, float* __restrict__ C,
                            int M, int N, int K, int totalTiles) {
    const int wave = threadIdx.x >> 5;
    const int lane = threadIdx.x & 31;
    const int tile = blockIdx.x * (blockDim.x >> 5) + wave;
    if (tile >= totalTiles) return;                 // wave-uniform exit

    const int ntiles = (N + 15) >> 4;
    const int mt = tile / ntiles;
    const int nt = tile - mt * ntiles;

    const int l15 = lane & 15;
    const int kb  = (lane >> 4) << 1;               // 0 or 2: K sub-lane offset
    const int arow = (mt << 4) + l15;               // A row (always < M)
    const int ccol = (nt << 4) + l15;               // B / C column
    const bool colOK = ccol < N;
    const int  bcol  = colOK ? ccol : 0;

    const float* __restrict__ ap = A + (size_t)arow * K + kb;
    const float* __restrict__ bp = B + (size_t)kb * N + bcol;

    v8f acc = {};
    for (int k = 0; k < K; k += 4) {
        v2f a;
        a.x = ap[k];
        a.y = ap[k + 1];
        if (RELU_A) { a.x = fmaxf(a.x, 0.0f); a.y = fmaxf(a.y, 0.0f); }
        float b0 = bp[(size_t)k * N];
        float b1 = bp[(size_t)(k + 1) * N];
        v2f b;
        b.x = colOK ? b0 : 0.0f;
        b.y = colOK ? b1 : 0.0f;
        acc = __builtin_amdgcn_wmma_f32_16x16x4_f32(
            /*neg_a=*/false, a, /*neg_b=*/false, b,
            /*c_mod=*/(short)0, acc, /*reuse_a=*/false, /*reuse_b=*/false);
    }

    if (colOK) {
        const float bv = ADD_BIAS ? bias[ccol] : 0.0f;
        const int drow = (mt << 4) + ((lane >> 4) << 3);   // lanes 16-31 -> M+8
        float* __restrict__ cp = C + (size_t)drow * N + ccol;
#pragma unroll
        for (int v = 0; v < 8; ++v)
            cp[(size_t)v * N] = acc[v] + bv;
    }
}

// ---------------- aggregation ----------------
// init: out[i,f] = dis[i]^2 * xw[i,f] + bias[f]   (self-loop term + bias folded in)
__global__ void k_agg_init(const float* __restrict__ xw, const float* __restrict__ dis,
                           const float* __restrict__ bias, float* __restrict__ out,
                           int n, int F) {
    int idx = blockIdx.x * blockDim.x + threadIdx.x;
    if (idx >= n * F) return;
    int i = idx / F;
    int f = idx - i * F;
    float d = dis[i];
    out[idx] = d * d * xw[idx] + bias[f];
}

// edges: out[col,f..f+3] += nrm[e] * xw[row,f..f+3]   (F % 4 == 0)
__global__ void k_scatter(const int* __restrict__ row, const int* __restrict__ col,
                          const float* __restrict__ nrm, const float* __restrict__ xw,
                          float* __restrict__ out, int E, int F) {
    const int chunks = F >> 2;
    long long idx = (long long)blockIdx.x * blockDim.x + threadIdx.x;
    if (idx >= (long long)E * chunks) return;
    int e = (int)(idx / chunks);
    int f = (int)(idx - (long long)e * chunks) << 2;
    float nv = nrm[e];
    const float4 v = *(const float4*)(xw + (size_t)row[e] * F + f);
    float* __restrict__ dst = out + (size_t)col[e] * F + f;
    atomicAdd(dst + 0, nv * v.x);
    atomicAdd(dst + 1, nv * v.y);
    atomicAdd(dst + 2, nv * v.z);
    atomicAdd(dst + 3, nv * v.w);
}

// ---------------- launch ----------------

extern "C" void kernel_launch(void* const* d_in, const int* in_sizes, int n_in,
                              void* d_out, int out_size, void* d_ws, size_t ws_size,
                              hipStream_t stream) {
    const float* x   = (const float*)d_in[0];
    const int*   ei  = (const int*)d_in[1];
    const float* W1  = (const float*)d_in[2];
    const float* b1  = (const float*)d_in[3];
    const float* W2  = (const float*)d_in[4];
    const float* b2  = (const float*)d_in[5];
    const float* Wfc = (const float*)d_in[6];
    const float* bfc = (const float*)d_in[7];
    float* out = (float*)d_out;

    const int E = in_sizes[1] / 2;
    const int* row = ei;        // edge_index[0]
    const int* col = ei + E;    // edge_index[1]

    uintptr_t base = (uintptr_t)d_ws;
    auto take = [&](size_t nfloats) {
        float* p = (float*)base;
        base += (nfloats * sizeof(float) + 255) & ~(size_t)255;
        return p;
    };
    float* deg = take(NN);
    float* dis = take(NN);
    float* nrm = take((size_t)E);
    float* xw1 = take((size_t)NN * H1DIM);
    float* h1  = take((size_t)NN * H1DIM);
    float* xw2 = take((size_t)NN * H2DIM);
    float* h2  = take((size_t)NN * H2DIM);

    const int T = 256;
    auto cdiv = [](long long a, long long b) { return (int)((a + b - 1) / b); };

    // symmetric normalization (self-loops folded into deg init = 1.0)
    k_fill<<<cdiv(NN, T), T, 0, stream>>>(deg, 1.0f, NN);
    k_deg <<<cdiv(E, T), T, 0, stream>>>(col, deg, E);
    k_dis <<<cdiv(NN, T), T, 0, stream>>>(deg, dis, NN);
    k_norm<<<cdiv(E, T), T, 0, stream>>>(row, col, dis, nrm, E);

    // ---- layer 1: xw1 = x @ W1 ; h1 = scatter(norm * xw1) + dis^2*xw1 + b1
    {
        const int tiles = (NN / 16) * ((H1DIM + 15) / 16);
        k_gemm_wmma<false, false><<<cdiv(tiles, 4), 128, 0, stream>>>(
            x, W1, nullptr, xw1, NN, H1DIM, FIN, tiles);
        k_agg_init<<<cdiv((long long)NN * H1DIM, T), T, 0, stream>>>(xw1, dis, b1, h1, NN, H1DIM);
        k_scatter<<<cdiv((long long)E * (H1DIM / 4), T), T, 0, stream>>>(row, col, nrm, xw1, h1, E, H1DIM);
    }
    // ---- layer 2: xw2 = relu(h1) @ W2 ; h2 = scatter + self + b2
    {
        const int tiles = (NN / 16) * ((H2DIM + 15) / 16);
        k_gemm_wmma<true, false><<<cdiv(tiles, 4), 128, 0, stream>>>(
            h1, W2, nullptr, xw2, NN, H2DIM, H1DIM, tiles);
        k_agg_init<<<cdiv((long long)NN * H2DIM, T), T, 0, stream>>>(xw2, dis, b2, h2, NN, H2DIM);
        k_scatter<<<cdiv((long long)E * (H2DIM / 4), T), T, 0, stream>>>(row, col, nrm, xw2, h2, E, H2DIM);
    }
    // ---- fc: out = relu(h2) @ Wfc + bfc
    {
        const int tiles = (NN / 16) * (FOUT / 16);
        k_gemm_wmma<true, true><<<cdiv(tiles, 4), 128, 0, stream>>>(
            h2, Wfc, bfc, out, NN, FOUT, H2DIM, tiles);
    }
}